// GCN_15367392985478
// MI455X (gfx1250) — compile-verified
//
#include <hip/hip_runtime.h>
#include <math.h>

// ---------------------------------------------------------------------------
// GAT 3-layer forward for gfx1250 (MI455X).
// GEMMs via V_WMMA_F32_16X16X4_F32 (fp32 WMMA, wave32). Edge softmax +
// aggregation via global atomics (bandwidth-bound part of the workload).
// ---------------------------------------------------------------------------

typedef float v2f __attribute__((ext_vector_type(2)));
typedef float v8f __attribute__((ext_vector_type(8)));

#define NNODES   50000
#define LNEPS    1e-5f
#define SLOPE    0.2f

// ---- order-preserving float <-> uint map for atomic float max ----
__device__ __forceinline__ unsigned ordmap(float f) {
    unsigned u = __float_as_uint(f);
    return (u & 0x80000000u) ? ~u : (u | 0x80000000u);
}
__device__ __forceinline__ float ordunmap(unsigned u) {
    return __uint_as_float((u & 0x80000000u) ? (u & 0x7FFFFFFFu) : ~u);
}

__device__ __forceinline__ void edge_ends(const int* __restrict__ ei, int E,
                                          int e, int& s, int& d) {
    if (e < E) { s = ei[e]; d = ei[E + e]; }
    else       { s = e - E; d = e - E; }
}

// ---------------------------------------------------------------------------
// GEMM: Hout[M,Ncol] = X[M,K] @ W[K,Ncol]   (ldx==K, ldw==Ncol, ldh==Ncol)
// One wave per 16x16 tile, fp32 WMMA with K-step 4. All indexing is 32-bit
// off the kernarg base pointers so the backend can use GVS (saddr) vector
// loads instead of per-lane 64-bit pointer arithmetic.
//   A 16x4 frag : lanes 0-15 row M=l, v0=K0 v1=K1 ; lanes 16-31 K2,K3
//   B 4x16 frag : lanes 0-15 v0=row K0 v1=row K1  ; lanes 16-31 rows K2,K3
//   C/D 16x16   : lanes 0-15 rows 0..7 ; lanes 16-31 rows 8..15
// ---------------------------------------------------------------------------
__global__ void gat_gemm_wmma(const float* __restrict__ X,
                              const float* __restrict__ W,
                              float* __restrict__ Hout,
                              int K, int Ncol, int rTiles, int cTiles) {
    int wv = (blockIdx.x * blockDim.x + threadIdx.x) >> 5;
    if (wv >= rTiles * cTiles) return;          // whole-wave uniform exit
    int lane = threadIdx.x & 31;
    int g = lane >> 4;          // lane group 0/1
    int l = lane & 15;
    int tr = wv / cTiles;
    int tc = wv % cTiles;

    // 32-bit offsets (max extent 50000*256 = 12.8M floats, fits easily)
    int xoff = (tr * 16 + l) * K + 2 * g;
    int woff = (2 * g) * Ncol + tc * 16 + l;

    v8f acc = {0.f, 0.f, 0.f, 0.f, 0.f, 0.f, 0.f, 0.f};
#pragma unroll 4
    for (int kb = 0; kb < K; kb += 4) {
        v2f a, b;
        a[0] = X[xoff + kb];
        a[1] = X[xoff + kb + 1];
        int wi = woff + kb * Ncol;
        b[0] = W[wi];
        b[1] = W[wi + Ncol];
        if ((kb & 15) == 0)                     // resolved statically under unroll
            __builtin_prefetch(&X[xoff + kb + 16], 0, 1);  // global_prefetch_b8
        acc = __builtin_amdgcn_wmma_f32_16x16x4_f32(
                  false, a, false, b, (short)0, acc, false, false);
    }

    int ooff = (tr * 16 + 8 * g) * Ncol + tc * 16 + l;
#pragma unroll
    for (int r = 0; r < 8; ++r) Hout[ooff + r * Ncol] = acc[r];
}

// ---------------------------------------------------------------------------
// Zero-pad W3 [96,40] -> [96,48]
// ---------------------------------------------------------------------------
__global__ void gat_pad_w3(const float* __restrict__ W3, float* __restrict__ W3p) {
    int gid = blockIdx.x * blockDim.x + threadIdx.x;
    if (gid >= 96 * 48) return;
    int r = gid / 48, c = gid % 48;
    W3p[gid] = (c < 40) ? W3[r * 40 + c] : 0.f;
}

// ---------------------------------------------------------------------------
// Per-node attention dots: s[n,h] = <h[n,h,:], a_src[h,:]> ; t likewise.
// ---------------------------------------------------------------------------
__global__ void gat_st(const float* __restrict__ Hf, int ldh,
                       const float* __restrict__ as_, const float* __restrict__ ad_,
                       int Hh, int D,
                       float* __restrict__ S, float* __restrict__ T, int total) {
    int gid = blockIdx.x * blockDim.x + threadIdx.x;
    if (gid >= total) return;
    int n = gid / Hh, h = gid % Hh;
    const float* hp = Hf + (size_t)n * ldh + h * D;
    const float* ap = as_ + h * D;
    const float* bp = ad_ + h * D;
    float s = 0.f, t = 0.f;
    for (int d = 0; d < D; ++d) { float v = hp[d]; s += v * ap[d]; t += v * bp[d]; }
    S[gid] = s;
    T[gid] = t;
}

__global__ void gat_init_mz(unsigned* __restrict__ M, float* __restrict__ Z, int total) {
    int gid = blockIdx.x * blockDim.x + threadIdx.x;
    if (gid >= total) return;
    M[gid] = ordmap(-3.0e38f);
    Z[gid] = 0.f;
}

__global__ void gat_init_out(float* __restrict__ O, const float* __restrict__ bias,
                             int Hd, int total) {
    int gid = blockIdx.x * blockDim.x + threadIdx.x;
    if (gid >= total) return;
    O[gid] = bias[gid % Hd];
}

// ---------------------------------------------------------------------------
// Edge pass A: segment max of leaky_relu(s[src]+t[dst]) per (dst, head)
// ---------------------------------------------------------------------------
__global__ void gat_edge_max(const int* __restrict__ ei, int E, int EE, int Hh,
                             const float* __restrict__ S, const float* __restrict__ T,
                             unsigned* __restrict__ M, int total) {
    int gid = blockIdx.x * blockDim.x + threadIdx.x;
    if (gid >= total) return;
    int e = gid / Hh, h = gid % Hh;
    int s, d; edge_ends(ei, E, e, s, d);
    float l = S[s * Hh + h] + T[d * Hh + h];
    l = (l > 0.f) ? l : SLOPE * l;
    atomicMax(&M[d * Hh + h], ordmap(l));
}

// ---------------------------------------------------------------------------
// Edge pass B: z[dst,h] += exp(e - m[dst,h])
// ---------------------------------------------------------------------------
__global__ void gat_edge_sum(const int* __restrict__ ei, int E, int EE, int Hh,
                             const float* __restrict__ S, const float* __restrict__ T,
                             const unsigned* __restrict__ M, float* __restrict__ Z,
                             int total) {
    int gid = blockIdx.x * blockDim.x + threadIdx.x;
    if (gid >= total) return;
    int e = gid / Hh, h = gid % Hh;
    int s, d; edge_ends(ei, E, e, s, d);
    float l = S[s * Hh + h] + T[d * Hh + h];
    l = (l > 0.f) ? l : SLOPE * l;
    int idx = d * Hh + h;
    atomicAdd(&Z[idx], __expf(l - ordunmap(M[idx])));
}

// ---------------------------------------------------------------------------
// Edge pass C: out[dst, c] += alpha(e,h) * h[src, c]   (c = h*D + d)
// One thread per (edge, feature) -> coalesced gather/scatter rows.
// ---------------------------------------------------------------------------
__global__ void gat_edge_agg(const int* __restrict__ ei, int E, int EE,
                             int Hh, int D, int Hd, int ldh, int ldo,
                             const float* __restrict__ S, const float* __restrict__ T,
                             const unsigned* __restrict__ M, const float* __restrict__ Z,
                             const float* __restrict__ Hf, float* __restrict__ O,
                             int total) {
    int gid = blockIdx.x * blockDim.x + threadIdx.x;
    if (gid >= total) return;
    int e = gid / Hd, c = gid % Hd;
    int h = c / D;
    int s, d; edge_ends(ei, E, e, s, d);
    float l = S[s * Hh + h] + T[d * Hh + h];
    l = (l > 0.f) ? l : SLOPE * l;
    int idx = d * Hh + h;
    float alpha = __expf(l - ordunmap(M[idx])) / (Z[idx] + 1e-16f);
    atomicAdd(&O[(size_t)d * ldo + c], alpha * Hf[(size_t)s * ldh + c]);
}

// ---------------------------------------------------------------------------
// LayerNorm + ELU, one wave32 per row. Writes Xout[row, 0..Hd)
// ---------------------------------------------------------------------------
__global__ void gat_ln_elu(const float* __restrict__ O, int Hd,
                           const float* __restrict__ lnw, const float* __restrict__ lnb,
                           float* __restrict__ Xout, int rows) {
    int wv = (blockIdx.x * blockDim.x + threadIdx.x) >> 5;
    if (wv >= rows) return;
    int lane = threadIdx.x & 31;
    const float* op = O + (size_t)wv * Hd;
    float sum = 0.f, sq = 0.f;
    for (int c = lane; c < Hd; c += 32) { float v = op[c]; sum += v; sq += v * v; }
    for (int off = 16; off > 0; off >>= 1) {
        sum += __shfl_down(sum, off, 32);
        sq  += __shfl_down(sq,  off, 32);
    }
    sum = __shfl(sum, 0, 32);
    sq  = __shfl(sq,  0, 32);
    float inv = 1.f / (float)Hd;
    float mu  = sum * inv;
    float var = sq * inv - mu * mu;
    float rs  = rsqrtf(var + LNEPS);
    float* xp = Xout + (size_t)wv * Hd;
    for (int c = lane; c < Hd; c += 32) {
        float y = (op[c] - mu) * rs * lnw[c] + lnb[c];
        xp[c] = (y > 0.f) ? y : (__expf(y) - 1.f);
    }
}

// ---------------------------------------------------------------------------
// Layer 3 finalize: LayerNorm(40) + log_softmax(40), one wave32 per row.
// ---------------------------------------------------------------------------
__global__ void gat_ln_logsoftmax(const float* __restrict__ O,
                                  const float* __restrict__ lnw,
                                  const float* __restrict__ lnb,
                                  float* __restrict__ out, int rows) {
    const int C = 40;
    int wv = (blockIdx.x * blockDim.x + threadIdx.x) >> 5;
    if (wv >= rows) return;
    int lane = threadIdx.x & 31;
    const float* op = O + (size_t)wv * C;
    int c0 = lane, c1 = lane + 32;
    bool v1ok = (c1 < C);
    float a0 = op[c0];
    float a1 = v1ok ? op[c1] : 0.f;
    float sum = a0 + a1;
    float sq  = a0 * a0 + a1 * a1;
    for (int off = 16; off > 0; off >>= 1) {
        sum += __shfl_down(sum, off, 32);
        sq  += __shfl_down(sq,  off, 32);
    }
    sum = __shfl(sum, 0, 32);
    sq  = __shfl(sq,  0, 32);
    float inv = 1.f / (float)C;
    float mu  = sum * inv;
    float var = sq * inv - mu * mu;
    float rs  = rsqrtf(var + LNEPS);
    float y0 = (a0 - mu) * rs * lnw[c0] + lnb[c0];
    float y1 = v1ok ? ((a1 - mu) * rs * lnw[c1] + lnb[c1]) : -3.0e38f;
    float mx = fmaxf(y0, y1);
    for (int off = 16; off > 0; off >>= 1) mx = fmaxf(mx, __shfl_down(mx, off, 32));
    mx = __shfl(mx, 0, 32);
    float se = __expf(y0 - mx) + (v1ok ? __expf(y1 - mx) : 0.f);
    for (int off = 16; off > 0; off >>= 1) se += __shfl_down(se, off, 32);
    se = __shfl(se, 0, 32);
    float L = mx + logf(se);
    float* outp = out + (size_t)wv * C;
    outp[c0] = y0 - L;
    if (v1ok) outp[c1] = y1 - L;
}

// ---------------------------------------------------------------------------
// Host side
// ---------------------------------------------------------------------------
static inline int gridFor(long long total) { return (int)((total + 255) / 256); }

extern "C" void kernel_launch(void* const* d_in, const int* in_sizes, int n_in,
                              void* d_out, int out_size, void* d_ws, size_t ws_size,
                              hipStream_t stream) {
    const float* x   = (const float*)d_in[0];
    const int*   ei  = (const int*)d_in[1];
    const float* W1  = (const float*)d_in[2];
    const float* a1s = (const float*)d_in[3];
    const float* a1d = (const float*)d_in[4];
    const float* b1  = (const float*)d_in[5];
    const float* l1w = (const float*)d_in[6];
    const float* l1b = (const float*)d_in[7];
    const float* W2  = (const float*)d_in[8];
    const float* a2s = (const float*)d_in[9];
    const float* a2d = (const float*)d_in[10];
    const float* b2  = (const float*)d_in[11];
    const float* l2w = (const float*)d_in[12];
    const float* l2b = (const float*)d_in[13];
    const float* W3  = (const float*)d_in[14];
    const float* a3s = (const float*)d_in[15];
    const float* a3d = (const float*)d_in[16];
    const float* b3  = (const float*)d_in[17];
    const float* l3w = (const float*)d_in[18];
    const float* l3b = (const float*)d_in[19];

    const int N  = NNODES;
    const int E  = in_sizes[1] / 2;
    const int EE = E + N;

    // workspace layout (floats)
    float*    ws   = (float*)d_ws;
    const size_t SH = (size_t)N * 192;
    float*    hbuf = ws;                 // [N,192] max
    float*    obuf = ws + SH;            // [N,192] max
    float*    xbuf = ws + 2 * SH;        // [N,192] max
    float*    sbuf = ws + 3 * SH;        // [N,6] max
    float*    tbuf = sbuf + (size_t)N * 6;
    unsigned* mbuf = (unsigned*)(tbuf + (size_t)N * 6);
    float*    zbuf = (float*)(mbuf + (size_t)N * 6);
    float*    w3p  = zbuf + (size_t)N * 6;   // [96,48]

    const int rT = N / 16;  // 3125

    // ---- pad W3 ----
    gat_pad_w3<<<gridFor(96 * 48), 256, 0, stream>>>(W3, w3p);

    // ================= Layer 1: IN=256 -> H=6, D=32, Hd=192 =================
    {
        const int K = 256, Hd = 192, Hh = 6, D = 32, cT = Hd / 16;
        gat_gemm_wmma<<<gridFor((long long)rT * cT * 32), 256, 0, stream>>>(
            x, W1, hbuf, K, Hd, rT, cT);
        gat_st<<<gridFor((long long)N * Hh), 256, 0, stream>>>(
            hbuf, Hd, a1s, a1d, Hh, D, sbuf, tbuf, N * Hh);
        gat_init_mz<<<gridFor((long long)N * Hh), 256, 0, stream>>>(mbuf, zbuf, N * Hh);
        gat_init_out<<<gridFor((long long)N * Hd), 256, 0, stream>>>(obuf, b1, Hd, N * Hd);
        gat_edge_max<<<gridFor((long long)EE * Hh), 256, 0, stream>>>(
            ei, E, EE, Hh, sbuf, tbuf, mbuf, EE * Hh);
        gat_edge_sum<<<gridFor((long long)EE * Hh), 256, 0, stream>>>(
            ei, E, EE, Hh, sbuf, tbuf, mbuf, zbuf, EE * Hh);
        gat_edge_agg<<<gridFor((long long)EE * Hd), 256, 0, stream>>>(
            ei, E, EE, Hh, D, Hd, Hd, Hd, sbuf, tbuf, mbuf, zbuf, hbuf, obuf, EE * Hd);
        gat_ln_elu<<<gridFor((long long)N * 32), 256, 0, stream>>>(
            obuf, Hd, l1w, l1b, xbuf, N);
    }

    // ================= Layer 2: 192 -> H=3, D=32, Hd=96 =====================
    {
        const int K = 192, Hd = 96, Hh = 3, D = 32, cT = Hd / 16;
        gat_gemm_wmma<<<gridFor((long long)rT * cT * 32), 256, 0, stream>>>(
            xbuf, W2, hbuf, K, Hd, rT, cT);
        gat_st<<<gridFor((long long)N * Hh), 256, 0, stream>>>(
            hbuf, Hd, a2s, a2d, Hh, D, sbuf, tbuf, N * Hh);
        gat_init_mz<<<gridFor((long long)N * Hh), 256, 0, stream>>>(mbuf, zbuf, N * Hh);
        gat_init_out<<<gridFor((long long)N * Hd), 256, 0, stream>>>(obuf, b2, Hd, N * Hd);
        gat_edge_max<<<gridFor((long long)EE * Hh), 256, 0, stream>>>(
            ei, E, EE, Hh, sbuf, tbuf, mbuf, EE * Hh);
        gat_edge_sum<<<gridFor((long long)EE * Hh), 256, 0, stream>>>(
            ei, E, EE, Hh, sbuf, tbuf, mbuf, zbuf, EE * Hh);
        gat_edge_agg<<<gridFor((long long)EE * Hd), 256, 0, stream>>>(
            ei, E, EE, Hh, D, Hd, Hd, Hd, sbuf, tbuf, mbuf, zbuf, hbuf, obuf, EE * Hd);
        gat_ln_elu<<<gridFor((long long)N * 32), 256, 0, stream>>>(
            obuf, Hd, l2w, l2b, xbuf, N);
    }

    // ============ Layer 3: 96 -> H=1, D=40 (GEMM padded to 48) ==============
    {
        const int K = 96, Np = 48, Hd = 40, Hh = 1, D = 40, cT = Np / 16;
        gat_gemm_wmma<<<gridFor((long long)rT * cT * 32), 256, 0, stream>>>(
            xbuf, w3p, hbuf, K, Np, rT, cT);                       // hbuf ld=48
        gat_st<<<gridFor((long long)N * Hh), 256, 0, stream>>>(
            hbuf, Np, a3s, a3d, Hh, D, sbuf, tbuf, N * Hh);
        gat_init_mz<<<gridFor((long long)N * Hh), 256, 0, stream>>>(mbuf, zbuf, N * Hh);
        gat_init_out<<<gridFor((long long)N * Hd), 256, 0, stream>>>(obuf, b3, Hd, N * Hd);
        gat_edge_max<<<gridFor((long long)EE * Hh), 256, 0, stream>>>(
            ei, E, EE, Hh, sbuf, tbuf, mbuf, EE * Hh);
        gat_edge_sum<<<gridFor((long long)EE * Hh), 256, 0, stream>>>(
            ei, E, EE, Hh, sbuf, tbuf, mbuf, zbuf, EE * Hh);
        gat_edge_agg<<<gridFor((long long)EE * Hd), 256, 0, stream>>>(
            ei, E, EE, Hh, D, Hd, Np, Hd, sbuf, tbuf, mbuf, zbuf, hbuf, obuf, EE * Hd);
        gat_ln_logsoftmax<<<gridFor((long long)N * 32), 256, 0, stream>>>(
            obuf, l3w, l3b, (float*)d_out, N);
    }
}